// RefineDetLoss_74869869904005
// MI455X (gfx1250) — compile-verified
//
#include <hip/hip_runtime.h>
#include <stdint.h>

// Problem dims (fixed by the reference setup_inputs()).
#define BB 32          // batch
#define AA 16320       // anchors
#define CC 21          // classes
#define MM 20          // GT boxes per image
#define NBLK 32        // anchor-chunk blocks per batch (each covers 512 anchors)
#define SORTN 16384    // next pow2 >= AA, bitonic sort size
#define NEG_INF_F (-1.0e30f)

typedef float        v2f __attribute__((ext_vector_type(2)));
typedef float        v8f __attribute__((ext_vector_type(8)));
typedef unsigned int v4u __attribute__((ext_vector_type(4)));
typedef int          v8i __attribute__((ext_vector_type(8)));
typedef int          v4i __attribute__((ext_vector_type(4)));

__device__ __forceinline__ float smoothl1(float x) {
  float a = fabsf(x);
  return a < 1.0f ? 0.5f * a * a : a - 0.5f;
}

__device__ __forceinline__ float iou_pf(float gx1, float gy1, float gx2, float gy2, float garea,
                                        float x1, float y1, float x2, float y2) {
  float ix1 = fmaxf(gx1, x1), iy1 = fmaxf(gy1, y1);
  float ix2 = fminf(gx2, x2), iy2 = fminf(gy2, y2);
  float iw = fmaxf(ix2 - ix1, 0.0f), ih = fmaxf(iy2 - iy1, 0.0f);
  float inter = iw * ih;
  float area = (x2 - x1) * (y2 - y1);
  return inter / (garea + area - inter);
}

// ---- CDNA5 Tensor Data Mover: DMA a contiguous run of f32 from global to LDS.
// D# per ISA ch.8: group0 {count=1, lds_addr, global_addr, type=2},
// group1 {data_size=4B, tensor_dim0=n, tensor_dim1=1, tile_dim0=n, tile_dim1=1,
//         tensor_dim0_stride=n}. Groups 2/3 unused (<=2-D tensor).
// amdgpu-toolchain (clang-23) 6-arg builtin form.
__device__ __forceinline__ void stage_lds_f32(float* dst_lds, const float* __restrict__ src,
                                              int n, int tid, int nthr) {
#if defined(__gfx1250__) && __has_builtin(__builtin_amdgcn_tensor_load_to_lds) && __has_builtin(__builtin_amdgcn_s_wait_tensorcnt)
  if (tid < 32) {  // wave-uniform branch: one wave issues the descriptor
    unsigned long long ga = (unsigned long long)(uintptr_t)src;
    unsigned lds = (unsigned)(uintptr_t)dst_lds;  // flat LDS addr: low 32 bits = byte offset
    v4u g0 = {0u, 0u, 0u, 0u};
    g0[0] = 1u;                                                    // count=1 (user D#)
    g0[1] = lds;                                                   // lds_addr
    g0[2] = (unsigned)(ga & 0xffffffffull);                        // global_addr[31:0]
    g0[3] = (unsigned)((ga >> 32) & 0x01ffffffull) | (2u << 30);   // global_addr[56:32] | type=2
    v8i g1 = {0, 0, 0, 0, 0, 0, 0, 0};
    unsigned un = (unsigned)n;
    g1[0] = (int)(2u << 16);                                       // data_size = 4 bytes
    g1[1] = (int)((un & 0xffffu) << 16);                           // tensor_dim0[15:0] @ bits 63:48
    g1[2] = (int)(((un >> 16) & 0xffffu) | (1u << 16));            // tensor_dim0[31:16] | tensor_dim1=1
    g1[3] = (int)((un & 0xffffu) << 16);                           // tile_dim0 @ bits 127:112
    g1[4] = 1;                                                     // tile_dim1 = 1
    g1[5] = (int)un;                                               // tensor_dim0_stride[31:0]
    v4i z4 = {0, 0, 0, 0};
    v8i z8 = {0, 0, 0, 0, 0, 0, 0, 0};
    __builtin_amdgcn_tensor_load_to_lds(g0, g1, z4, z4, z8, 0);
    __builtin_amdgcn_s_wait_tensorcnt(0);
  }
#else
  for (int i = tid; i < n; i += nthr) dst_lds[i] = src[i];
#endif
}

// ---- Block sum: LDS tree down to 64 partials, then a single
// V_WMMA_F32_16X16X4_F32 with A = ones(16x4) folds 64 -> 16 column sums on the
// matrix pipe (exact: sum over the 16 columns of D == sum of all 64 B cells,
// and lanes 0..15 VGPR0 hold D[0][0..15] per the documented C/D layout).
__device__ __forceinline__ float block_reduce_sum(float v, float* sm) {
  const int tid = threadIdx.x;
  const int nthr = blockDim.x;
  __syncthreads();           // buffer may be in use by caller
  sm[tid] = v;
  __syncthreads();
  for (int s = nthr >> 1; s >= 64; s >>= 1) {
    if (tid < s) sm[tid] += sm[tid + s];
    __syncthreads();
  }
#if defined(__gfx1250__) && __has_builtin(__builtin_amdgcn_wmma_f32_16x16x4_f32)
  if (tid < 32) {            // whole wave 0: EXEC all ones inside
    v2f a; a[0] = 1.0f; a[1] = 1.0f;             // A = ones(16x4)
    v2f b; b[0] = sm[tid]; b[1] = sm[tid + 32];  // 64 partials fill B(4x16)
    v8f c = {};
    c = __builtin_amdgcn_wmma_f32_16x16x4_f32(false, a, false, b, (short)0, c, false, false);
    if (tid < 16) sm[tid] = c[0];                // D[0][j], j = lane
  }
  __syncthreads();
  if (tid == 0) {
    float s = 0.0f;
    for (int i = 0; i < 16; ++i) s += sm[i];
    sm[0] = s;
  }
#else
  if (tid < 32) sm[tid] += sm[tid + 32];
  __syncthreads();
  if (tid == 0) {
    float s = 0.0f;
    for (int i = 0; i < 32; ++i) s += sm[i];
    sm[0] = s;
  }
#endif
  __syncthreads();
  float r = sm[0];
  __syncthreads();
  return r;
}

// ---------------------------------------------------------------------------
__global__ void k0_init(float* acc, float* posCnt) {
  int t = threadIdx.x;
  if (t < 8) acc[t] = 0.0f;
  if (t < BB) posCnt[t] = 0.0f;
}

// k1: per-anchor best-GT IoU/idx for both stages + per-block per-GT argmax partials.
__global__ void k1_match(const float4* __restrict__ refine_loc,
                         const float4* __restrict__ anchors,
                         const float* __restrict__ gt_boxes,
                         float* __restrict__ ov1, int* __restrict__ id1,
                         float* __restrict__ ov2, int* __restrict__ id2,
                         float* __restrict__ pargm) {
  const int b = blockIdx.x, cb = blockIdx.y, tid = threadIdx.x;
  __shared__ float sgt[MM * 4];
  __shared__ float smv[256];
  __shared__ int smi[256];
  stage_lds_f32(sgt, gt_boxes + (size_t)b * MM * 4, MM * 4, tid, blockDim.x);
  __syncthreads();

  int aidx[2];
  float ax1[2], ay1[2], ax2[2], ay2[2];   // fixed anchors, point form
  float dx1[2], dy1[2], dx2[2], dy2[2];   // ARM-refined anchors, point form
  float bo1[2]; int bi1[2]; float bo2[2]; int bi2[2];

#pragma unroll
  for (int k = 0; k < 2; ++k) {
    int a = cb * 512 + k * 256 + tid;
    aidx[k] = a;
    bo1[k] = -1.0f; bi1[k] = 0; bo2[k] = -1.0f; bi2[k] = 0;
    if (a < AA) {
      float4 an = anchors[a];
      ax1[k] = an.x - 0.5f * an.z; ay1[k] = an.y - 0.5f * an.w;
      ax2[k] = an.x + 0.5f * an.z; ay2[k] = an.y + 0.5f * an.w;
      float4 rl = refine_loc[(size_t)b * AA + a];
      float dcx = an.x + rl.x * 0.1f * an.z;
      float dcy = an.y + rl.y * 0.1f * an.w;
      float dw = an.z * expf(rl.z * 0.2f);
      float dh = an.w * expf(rl.w * 0.2f);
      dx1[k] = dcx - 0.5f * dw; dy1[k] = dcy - 0.5f * dh;
      dx2[k] = dcx + 0.5f * dw; dy2[k] = dcy + 0.5f * dh;
    } else {
      ax1[k] = ay1[k] = ax2[k] = ay2[k] = 0.0f;
      dx1[k] = dy1[k] = dx2[k] = dy2[k] = 0.0f;
    }
  }

  for (int m = 0; m < MM; ++m) {
    float gx1 = sgt[m * 4 + 0], gy1 = sgt[m * 4 + 1];
    float gx2 = sgt[m * 4 + 2], gy2 = sgt[m * 4 + 3];
    float garea = (gx2 - gx1) * (gy2 - gy1);
    float c1 = NEG_INF_F; int c1i = 0x7fffffff;
    float c2 = NEG_INF_F; int c2i = 0x7fffffff;
#pragma unroll
    for (int k = 0; k < 2; ++k) {
      if (aidx[k] < AA) {
        float o1v = iou_pf(gx1, gy1, gx2, gy2, garea, ax1[k], ay1[k], ax2[k], ay2[k]);
        if (o1v > bo1[k]) { bo1[k] = o1v; bi1[k] = m; }    // first-m-wins == argmax(0)
        if (o1v > c1) { c1 = o1v; c1i = aidx[k]; }         // k ascending -> smaller a wins ties
        float o2v = iou_pf(gx1, gy1, gx2, gy2, garea, dx1[k], dy1[k], dx2[k], dy2[k]);
        if (o2v > bo2[k]) { bo2[k] = o2v; bi2[k] = m; }
        if (o2v > c2) { c2 = o2v; c2i = aidx[k]; }
      }
    }
    // stage-1 block argmax (smallest anchor index breaks ties, matching jnp.argmax)
    smv[tid] = c1; smi[tid] = c1i; __syncthreads();
    for (int s = 128; s > 0; s >>= 1) {
      if (tid < s) {
        if (smv[tid + s] > smv[tid] || (smv[tid + s] == smv[tid] && smi[tid + s] < smi[tid])) {
          smv[tid] = smv[tid + s]; smi[tid] = smi[tid + s];
        }
      }
      __syncthreads();
    }
    if (tid == 0) {
      float* p = pargm + ((((size_t)b * 2 + 0) * NBLK + cb) * MM + m) * 2;
      p[0] = smv[0]; p[1] = __int_as_float(smi[0]);
    }
    __syncthreads();
    // stage-2 block argmax
    smv[tid] = c2; smi[tid] = c2i; __syncthreads();
    for (int s = 128; s > 0; s >>= 1) {
      if (tid < s) {
        if (smv[tid + s] > smv[tid] || (smv[tid + s] == smv[tid] && smi[tid + s] < smi[tid])) {
          smv[tid] = smv[tid + s]; smi[tid] = smi[tid + s];
        }
      }
      __syncthreads();
    }
    if (tid == 0) {
      float* p = pargm + ((((size_t)b * 2 + 1) * NBLK + cb) * MM + m) * 2;
      p[0] = smv[0]; p[1] = __int_as_float(smi[0]);
    }
    __syncthreads();
  }

#pragma unroll
  for (int k = 0; k < 2; ++k) {
    if (aidx[k] < AA) {
      size_t o = (size_t)b * AA + aidx[k];
      ov1[o] = bo1[k]; id1[o] = bi1[k];
      ov2[o] = bo2[k]; id2[o] = bi2[k];
    }
  }
}

// k2: combine per-chunk argmax partials; apply the guaranteed-match override
// (sequential in m so duplicate best-anchors resolve last-GT-wins like the ref).
__global__ void k2_combine(float* __restrict__ ov1, int* __restrict__ id1,
                           float* __restrict__ ov2, int* __restrict__ id2,
                           const float* __restrict__ pargm) {
  const int b = blockIdx.x, tid = threadIdx.x;
  __shared__ int best[2][MM];
  if (tid < 2 * MM) {
    int s = tid / MM, m = tid % MM;
    float bv = NEG_INF_F; int bi = 0x7fffffff;
    for (int cb = 0; cb < NBLK; ++cb) {
      const float* p = pargm + ((((size_t)b * 2 + s) * NBLK + cb) * MM + m) * 2;
      float v = p[0]; int i = __float_as_int(p[1]);
      if (v > bv || (v == bv && i < bi)) { bv = v; bi = i; }
    }
    best[s][m] = bi;
  }
  __syncthreads();
  if (tid == 0) {
    for (int m = 0; m < MM; ++m) {
      size_t o = (size_t)b * AA + best[0][m];
      ov1[o] = 2.0f; id1[o] = m;
    }
  }
  if (tid == 1) {
    for (int m = 0; m < MM; ++m) {
      size_t o = (size_t)b * AA + best[1][m];
      ov2[o] = 2.0f; id2[o] = m;
    }
  }
}

// k3: fused ARM + ODM losses per anchor; emits neg_losses for mining.
__global__ void k3_losses(const float2* __restrict__ objectness,
                          const float4* __restrict__ refine_loc,
                          const float* __restrict__ pred_conf,
                          const float4* __restrict__ pred_loc,
                          const float4* __restrict__ anchors,
                          const float* __restrict__ gt_boxes,
                          const int* __restrict__ gt_labels,
                          const float* __restrict__ ov1, const int* __restrict__ id1,
                          const float* __restrict__ ov2, const int* __restrict__ id2,
                          float* __restrict__ negloss,
                          float* acc, float* posCnt) {
  const int b = blockIdx.x, cb = blockIdx.y, tid = threadIdx.x;
  __shared__ float sgt[MM * 4];
  __shared__ int slab[MM];
  __shared__ float sred[256];
  stage_lds_f32(sgt, gt_boxes + (size_t)b * MM * 4, MM * 4, tid, blockDim.x);
  if (tid < MM) slab[tid] = gt_labels[b * MM + tid];
  __syncthreads();

  float pArmN = 0.f, pArmLoc = 0.f, pArmCls = 0.f, pN2 = 0.f, pLoc2 = 0.f, pPosCE = 0.f;

  for (int k = 0; k < 2; ++k) {
    int a = cb * 512 + k * 256 + tid;
    if (a >= AA) continue;
    size_t o = (size_t)b * AA + a;
    float4 an = anchors[a];
    float4 rl = refine_loc[o];
    float2 ob = objectness[o];
    float o0 = ob.x, o1 = ob.y;
    float mx01 = fmaxf(o0, o1);
    float lse01 = mx01 + logf(expf(o0 - mx01) + expf(o1 - mx01));
    bool mask = (o1 - o0) > -4.59511985f;  // softmax(objectness)[0] < 0.99

    // ---- ARM stage ----
    float ov = ov1[o]; int gi = id1[o];
    int conf1 = (ov < 0.5f) ? 0 : slab[gi];
    float pos = (conf1 > 0) ? 1.0f : 0.0f;
    const float* g = &sgt[gi * 4];
    float t0 = ((g[0] + g[2]) * 0.5f - an.x) / (0.1f * an.z);
    float t1 = ((g[1] + g[3]) * 0.5f - an.y) / (0.1f * an.w);
    float t2 = logf((g[2] - g[0]) / an.z) / 0.2f;
    float t3 = logf((g[3] - g[1]) / an.w) / 0.2f;
    pArmLoc += pos * (smoothl1(rl.x - t0) + smoothl1(rl.y - t1) +
                      smoothl1(rl.z - t2) + smoothl1(rl.w - t3));
    pArmCls += lse01 - ((conf1 > 0) ? o1 : o0);
    pArmN += pos;

    // ---- ODM stage: decode refined anchor, re-encode vs. its matched GT ----
    float dcx = an.x + rl.x * 0.1f * an.z;
    float dcy = an.y + rl.y * 0.1f * an.w;
    float dw = an.z * expf(rl.z * 0.2f);
    float dh = an.w * expf(rl.w * 0.2f);
    float ovb = ov2[o]; int gi2 = id2[o];
    int conf2 = (ovb < 0.5f) ? 0 : slab[gi2];
    bool pos2 = (conf2 > 0) && mask;
    const float* g2 = &sgt[gi2 * 4];
    float u0 = ((g2[0] + g2[2]) * 0.5f - dcx) / (0.1f * dw);
    float u1 = ((g2[1] + g2[3]) * 0.5f - dcy) / (0.1f * dh);
    float u2 = logf((g2[2] - g2[0]) / dw) / 0.2f;
    float u3 = logf((g2[3] - g2[1]) / dh) / 0.2f;
    float4 pl = pred_loc[o];
    float slsum = smoothl1(pl.x - u0) + smoothl1(pl.y - u1) +
                  smoothl1(pl.z - u2) + smoothl1(pl.w - u3);
    pLoc2 += pos2 ? slsum : 0.0f;

    // 21-class CE; prefetch next iteration's row (global_prefetch_b8)
    const float* pc = pred_conf + o * CC;
    __builtin_prefetch(pc + (size_t)256 * CC, 0, 1);
    float cmx = pc[0];
#pragma unroll
    for (int j = 1; j < CC; ++j) cmx = fmaxf(cmx, pc[j]);
    float se = 0.0f;
#pragma unroll
    for (int j = 0; j < CC; ++j) se += expf(pc[j] - cmx);
    float ce2 = cmx + logf(se) - pc[conf2];
    pPosCE += pos2 ? ce2 : 0.0f;
    pN2 += pos2 ? 1.0f : 0.0f;
    negloss[o] = (mask && conf2 == 0) ? ce2 : NEG_INF_F;
  }

  float rArmN = block_reduce_sum(pArmN, sred);
  float rArmLoc = block_reduce_sum(pArmLoc, sred);
  float rArmCls = block_reduce_sum(pArmCls, sred);
  float rN2 = block_reduce_sum(pN2, sred);
  float rLoc2 = block_reduce_sum(pLoc2, sred);
  float rPosCE = block_reduce_sum(pPosCE, sred);
  if (tid == 0) {
    atomicAdd(&acc[0], rArmN);
    atomicAdd(&acc[1], rArmLoc);
    atomicAdd(&acc[2], rArmCls);
    atomicAdd(&acc[3], rN2);
    atomicAdd(&acc[4], rLoc2);
    atomicAdd(&acc[5], rPosCE);
    atomicAdd(&posCnt[b], rN2);
  }
}

// k4: per-batch hard-negative mining: bitonic sort (descending) in LDS, sum top-K.
__global__ void k4_topk(const float* __restrict__ negloss,
                        const float* __restrict__ posCnt,
                        float* acc) {
  const int b = blockIdx.x, tid = threadIdx.x;
  extern __shared__ float sml[];
  for (int i = tid; i < SORTN; i += blockDim.x)
    sml[i] = (i < AA) ? negloss[(size_t)b * AA + i] : -3.0e38f;
  __syncthreads();
  for (int k = 2; k <= SORTN; k <<= 1) {
    for (int j = k >> 1; j > 0; j >>= 1) {
      for (int i = tid; i < SORTN; i += blockDim.x) {
        int ixj = i ^ j;
        if (ixj > i) {
          float x = sml[i], y = sml[ixj];
          bool desc = ((i & k) == 0);
          if (desc ? (x < y) : (x > y)) { sml[i] = y; sml[ixj] = x; }
        }
      }
      __syncthreads();
    }
  }
  int pos_num = (int)(posCnt[b] + 0.5f);
  int neg_num = pos_num * 3;
  if (neg_num > AA - pos_num) neg_num = AA - pos_num;
  if (neg_num < 10) neg_num = 10;
  float part = 0.0f;
  for (int i = tid; i < neg_num; i += blockDim.x) {
    float v = sml[i];
    if (v > NEG_INF_F * 0.5f) part += v;   // sorted_neg > NEG_INF/2 filter
  }
  float tot = block_reduce_sum(part, sml); // safe: sorted data already consumed
  if (tid == 0) atomicAdd(&acc[6], tot);
}

__global__ void k5_final(const float* __restrict__ acc, float* __restrict__ out) {
  if (threadIdx.x == 0 && blockIdx.x == 0) {
    float armN = acc[0], armLoc = acc[1], armCls = acc[2];
    float N = acc[3], loc2 = acc[4], posCE = acc[5], negSum = acc[6];
    float class_loss = (posCE + negSum) / N;
    float loc_loss = loc2 / N;
    float arm_cls = 0.04f * armCls / armN;
    float arm_loc = armLoc / armN;
    out[0] = class_loss + loc_loss + arm_cls + arm_loc;
    out[1] = class_loss;
    out[2] = loc_loss;
    out[3] = arm_cls;
    out[4] = arm_loc;
  }
}

extern "C" void kernel_launch(void* const* d_in, const int* in_sizes, int n_in,
                              void* d_out, int out_size, void* d_ws, size_t ws_size,
                              hipStream_t stream) {
  const float2* objectness = (const float2*)d_in[0];
  const float4* refine_loc = (const float4*)d_in[1];
  const float* pred_conf = (const float*)d_in[2];
  const float4* pred_loc = (const float4*)d_in[3];
  const float4* anchors = (const float4*)d_in[4];
  const float* gt_boxes = (const float*)d_in[5];
  const int* gt_labels = (const int*)d_in[6];
  float* outf = (float*)d_out;

  char* p = (char*)d_ws;
  auto take = [&](size_t bytes) {
    char* r = p;
    p += (bytes + 255) & ~(size_t)255;
    return r;
  };
  float* ov1 = (float*)take((size_t)BB * AA * 4);
  int* id1 = (int*)take((size_t)BB * AA * 4);
  float* ov2 = (float*)take((size_t)BB * AA * 4);
  int* id2 = (int*)take((size_t)BB * AA * 4);
  float* pargm = (float*)take((size_t)BB * 2 * NBLK * MM * 2 * 4);
  float* negl = (float*)take((size_t)BB * AA * 4);
  float* posCnt = (float*)take((size_t)BB * 4);
  float* acc = (float*)take(8 * 4);

  k0_init<<<1, 64, 0, stream>>>(acc, posCnt);
  k1_match<<<dim3(BB, NBLK), 256, 0, stream>>>(refine_loc, anchors, gt_boxes,
                                               ov1, id1, ov2, id2, pargm);
  k2_combine<<<BB, 64, 0, stream>>>(ov1, id1, ov2, id2, pargm);
  k3_losses<<<dim3(BB, NBLK), 256, 0, stream>>>(objectness, refine_loc, pred_conf, pred_loc,
                                                anchors, gt_boxes, gt_labels,
                                                ov1, id1, ov2, id2, negl, acc, posCnt);
  k4_topk<<<BB, 1024, SORTN * sizeof(float), stream>>>(negl, posCnt, acc);
  k5_final<<<1, 1, 0, stream>>>(acc, outf);

  (void)in_sizes; (void)n_in; (void)out_size; (void)ws_size;
}